// MoE_14396730376790
// MI455X (gfx1250) — compile-verified
//
#include <hip/hip_runtime.h>

typedef __attribute__((ext_vector_type(2))) float v2f;
typedef __attribute__((ext_vector_type(8))) float v8f;

#define N_TOK   65536
#define D_INF   1024
#define D_OUTF  1024
#define NE      16
#define GRP     64                          // tokens per expert-GEMM group
#define LDS_STRIDE 1028                     // floats per A row (bank-conflict-free)
#define MAX_GROUPS (N_TOK / GRP + NE)       // 1040
#define SMEM_BYTES (GRP * LDS_STRIDE * 4 + GRP * 4)

__device__ __forceinline__ v8f wmma_f32_k4(v2f a, v2f b, v8f c) {
    // V_WMMA_F32_16X16X4_F32: D(16x16 f32) = A(16x4 f32) * B(4x16 f32) + C
    return __builtin_amdgcn_wmma_f32_16x16x4_f32(
        false, a, false, b, (short)0, c, false, false);
}

// --- kernel 0: zero expert counters -----------------------------------------
__global__ void moe_init(int* __restrict__ counts) {
    if (threadIdx.x < NE) counts[threadIdx.x] = 0;
}

// --- kernel 1: gating logits (WMMA) + argmax + histogram --------------------
// One wave per 16 tokens. logits[16 tok][16 exp] = x_tile @ Wg^T (+ bg).
__global__ __launch_bounds__(32) void moe_gate(
    const float* __restrict__ x, const float* __restrict__ Wg,
    const float* __restrict__ bg, int* __restrict__ idx,
    int* __restrict__ counts)
{
    const int lane = threadIdx.x;      // 0..31
    const int n    = lane & 15;        // A row m == B col n (expert)
    const int hi   = lane >> 4;
    const int koff = hi * 2;           // K pair selected by lane half
    const int tokBase = blockIdx.x * 16;

    const float* xa = x  + (size_t)(tokBase + n) * D_INF; // A: row m = n
    const float* wb = Wg + (size_t)n * D_INF;             // B: col n

    v8f c = {};
#pragma unroll 8
    for (int kb = 0; kb < D_INF; kb += 4) {
        v2f a = *(const v2f*)(xa + kb + koff);
        v2f b = *(const v2f*)(wb + kb + koff);
        c = wmma_f32_k4(a, b, c);
    }

    const float bias = bg[n];
#pragma unroll
    for (int v = 0; v < 8; ++v) {
        float best = c[v] + bias;      // logit for token (v + 8*hi), expert n
        int   bi   = n;
#pragma unroll
        for (int off = 8; off >= 1; off >>= 1) {   // argmax across 16 experts
            float ov = __shfl_xor(best, off, 16);
            int   oi = __shfl_xor(bi,   off, 16);
            if (ov > best || (ov == best && oi < bi)) { best = ov; bi = oi; }
        }
        if (n == 0) {                  // lanes 0 and 16 own disjoint tokens
            const int token = tokBase + v + 8 * hi;
            idx[token] = bi;
            atomicAdd(&counts[bi], 1);
        }
    }
}

// --- kernel 2: exclusive scan + group descriptors (trivial, single thread) --
__global__ void moe_scan(const int* __restrict__ counts, int* __restrict__ offsets,
                         int* __restrict__ cursor, int* __restrict__ desc)
{
    if (threadIdx.x != 0 || blockIdx.x != 0) return;
    int acc = 0;
    for (int e = 0; e < NE; ++e) {
        offsets[e] = acc; cursor[e] = acc; acc += counts[e];
    }
    offsets[NE] = acc;
    int g = 0;
    for (int e = 0; e < NE; ++e)
        for (int s = offsets[e]; s < offsets[e + 1]; s += GRP) {
            desc[2 * g] = e; desc[2 * g + 1] = s; ++g;
        }
    for (; g < MAX_GROUPS; ++g) { desc[2 * g] = -1; desc[2 * g + 1] = 0; }
}

// --- kernel 3: scatter token ids into per-expert buckets --------------------
__global__ __launch_bounds__(256) void moe_scatter(
    const int* __restrict__ idx, int* __restrict__ cursor, int* __restrict__ bucket)
{
    const int t = blockIdx.x * 256 + threadIdx.x;
    const int e = idx[t];
    const int p = atomicAdd(&cursor[e], 1);
    bucket[p] = t;
}

// --- kernel 4: expert GEMM on gathered 64-token groups (WMMA f32) -----------
// Block = 256 threads (8 waves). A tile: 64 tokens x 1024 f32 in LDS, loaded
// with CDNA5 async global->LDS (ASYNCcnt). Each wave owns 8 16-col tiles and
// 4 row tiles; one B fragment load feeds 4 WMMAs.
__global__ __launch_bounds__(256) void moe_expert(
    const float* __restrict__ x, const float* __restrict__ We,
    const float* __restrict__ be, const int* __restrict__ bucket,
    const int* __restrict__ offsets, const int* __restrict__ desc,
    float* __restrict__ out)
{
    extern __shared__ char smem[];
    float* ldsA = (float*)smem;                        // [GRP][LDS_STRIDE]
    int*   tok  = (int*)(smem + GRP * LDS_STRIDE * 4); // [GRP]

    const int e = desc[2 * blockIdx.x];
    if (e < 0) return;
    const int start = desc[2 * blockIdx.x + 1];
    const int rem   = offsets[e + 1] - start;
    const int cnt   = rem < GRP ? rem : GRP;

    for (int r = threadIdx.x; r < GRP; r += 256)
        tok[r] = (r < cnt) ? bucket[start + r] : -1;
    __syncthreads();

    // Gather 64 x 1024 f32 A-tile into LDS. Valid rows: async B128 copies
    // (global_load_async_to_lds_b128, tracked on ASYNCcnt); pad rows: ds zero.
    for (int i = 0; i < GRP; ++i) {
        float* dstp = ldsA + i * LDS_STRIDE + 4 * threadIdx.x;
        const int tk = tok[i];
        if (tk >= 0) {
            const float* gp = x + (size_t)tk * D_INF + 4 * threadIdx.x;
            const unsigned ldsAddr = (unsigned)(size_t)dstp; // low 32 bits = LDS byte addr
            asm volatile("global_load_async_to_lds_b128 %0, %1, off"
                         :: "v"(ldsAddr), "v"(gp) : "memory");
        } else {
            *(float4*)dstp = make_float4(0.f, 0.f, 0.f, 0.f);
        }
    }
    asm volatile("s_wait_asynccnt 0x0" ::: "memory");
    __syncthreads();

    const int lane = threadIdx.x & 31;
    const int wave = threadIdx.x >> 5;     // 8 waves x 8 col tiles = 1024 cols
    const int n    = lane & 15;            // A row m == B/D col n
    const int hi   = lane >> 4;
    const int koff = hi * 2;

    const float* a0 = ldsA + ( 0 + n) * LDS_STRIDE;
    const float* a1 = ldsA + (16 + n) * LDS_STRIDE;
    const float* a2 = ldsA + (32 + n) * LDS_STRIDE;
    const float* a3 = ldsA + (48 + n) * LDS_STRIDE;

    for (int ct = 0; ct < 8; ++ct) {
        const int colBase = (wave * 8 + ct) * 16;
        const float* Brow = We + ((size_t)e * D_OUTF + colBase + n) * D_INF;
        __builtin_prefetch(Brow, 0, 0);

        v8f c0 = {}, c1 = {}, c2 = {}, c3 = {};
#pragma unroll 4
        for (int kb = 0; kb < D_INF; kb += 4) {
            const v2f b  = *(const v2f*)(Brow + kb + koff);   // shared by 4 WMMAs
            const v2f f0 = *(const v2f*)(a0 + kb + koff);
            const v2f f1 = *(const v2f*)(a1 + kb + koff);
            const v2f f2 = *(const v2f*)(a2 + kb + koff);
            const v2f f3 = *(const v2f*)(a3 + kb + koff);
            c0 = wmma_f32_k4(f0, b, c0);
            c1 = wmma_f32_k4(f1, b, c1);
            c2 = wmma_f32_k4(f2, b, c2);
            c3 = wmma_f32_k4(f3, b, c3);
        }

        const float bias = be[(size_t)e * D_OUTF + colBase + n];
#pragma unroll
        for (int v = 0; v < 8; ++v) {
            const int row = v + 8 * hi;            // C VGPR v -> row v + 8*hi
            const int g0 = row, g1 = 16 + row, g2 = 32 + row, g3 = 48 + row;
            if (g0 < cnt) out[(size_t)tok[g0] * D_OUTF + colBase + n] = c0[v] + bias;
            if (g1 < cnt) out[(size_t)tok[g1] * D_OUTF + colBase + n] = c1[v] + bias;
            if (g2 < cnt) out[(size_t)tok[g2] * D_OUTF + colBase + n] = c2[v] + bias;
            if (g3 < cnt) out[(size_t)tok[g3] * D_OUTF + colBase + n] = c3[v] + bias;
        }
    }
}

extern "C" void kernel_launch(void* const* d_in, const int* in_sizes, int n_in,
                              void* d_out, int out_size, void* d_ws, size_t ws_size,
                              hipStream_t stream)
{
    const float* x  = (const float*)d_in[0];
    const float* Wg = (const float*)d_in[1];
    const float* bg = (const float*)d_in[2];
    const float* We = (const float*)d_in[3];
    const float* be = (const float*)d_in[4];
    float* out = (float*)d_out;

    int* ws      = (int*)d_ws;
    int* counts  = ws;                 // 16
    int* offsets = ws + 16;            // 17 (padded to 32)
    int* cursor  = ws + 48;            // 16
    int* idx     = ws + 64;            // N_TOK
    int* bucket  = idx + N_TOK;        // N_TOK
    int* desc    = bucket + N_TOK;     // 2 * MAX_GROUPS

    hipLaunchKernelGGL(moe_init,    dim3(1),           dim3(32),  0, stream, counts);
    hipLaunchKernelGGL(moe_gate,    dim3(N_TOK / 16),  dim3(32),  0, stream, x, Wg, bg, idx, counts);
    hipLaunchKernelGGL(moe_scan,    dim3(1),           dim3(32),  0, stream, counts, offsets, cursor, desc);
    hipLaunchKernelGGL(moe_scatter, dim3(N_TOK / 256), dim3(256), 0, stream, idx, cursor, bucket);
    hipLaunchKernelGGL(moe_expert,  dim3(MAX_GROUPS),  dim3(256), SMEM_BYTES, stream,
                       x, We, be, bucket, offsets, desc, out);
}